// SelfConnectionLinearIntro_35227321762139
// MI455X (gfx1250) — compile-verified
//
#include <hip/hip_runtime.h>

typedef __attribute__((ext_vector_type(2))) float v2f;
typedef __attribute__((ext_vector_type(8))) float v8f;

#define D0 128           // MUL0*1
#define D1 192           // MUL1*3
#define D2 160           // MUL2*5
#define DTOT 480

// LDS: weights pre-swizzled into WMMA B-fragment order.
// Fragment (kc, nt) occupies 64 floats: lane l holds {w[k][n], w[k+1][n]}
// at frag_base + l*2, with k = kc*4 + 2*(l>>4), n = nt*16 + (l&15).
// B0: kc in [0,32), nt in [0,8)  -> 32*8*64  = 16384 floats
// B1: kc in [0,16), nt in [0,4)  -> 16*4*64  =  4096 floats
// B2: kc in [0, 8), nt in [0,2)  ->  8*2*64  =  1024 floats
#define B0_OFF 0
#define B1_OFF 16384
#define B2_OFF (16384 + 4096)
#define LDS_FLOATS (16384 + 4096 + 1024)   // 21504 floats = 86016 bytes

__global__ __launch_bounds__(256)
void sc_linear_wmma_f32(const float* __restrict__ x,
                        const float* __restrict__ w0,
                        const float* __restrict__ w1,
                        const float* __restrict__ w2,
                        float* __restrict__ out,
                        int n_atoms)
{
    extern __shared__ float lds[];
    const int t = threadIdx.x;

    // ---- Stage weights into LDS, swizzling to B-fragment layout.
    // Source-linear iteration keeps global reads coalesced.
    for (int i = t; i < 128 * 128; i += 256) {
        const int k = i >> 7, c = i & 127;
        const int nt = c >> 4, n = c & 15;
        const int kc = k >> 2, kr = k & 3;
        const int lane = (kr >> 1) * 16 + n;
        lds[B0_OFF + ((kc * 8 + nt) * 32 + lane) * 2 + (kr & 1)] = w0[i];
    }
    for (int i = t; i < 64 * 64; i += 256) {
        const int k = i >> 6, c = i & 63;
        const int nt = c >> 4, n = c & 15;
        const int kc = k >> 2, kr = k & 3;
        const int lane = (kr >> 1) * 16 + n;
        lds[B1_OFF + ((kc * 4 + nt) * 32 + lane) * 2 + (kr & 1)] = w1[i];
    }
    for (int i = t; i < 32 * 32; i += 256) {
        const int k = i >> 5, c = i & 31;
        const int nt = c >> 4, n = c & 15;
        const int kc = k >> 2, kr = k & 3;
        const int lane = (kr >> 1) * 16 + n;
        lds[B2_OFF + ((kc * 2 + nt) * 32 + lane) * 2 + (kr & 1)] = w2[i];
    }
    __syncthreads();

    const int wave = t >> 5;
    const int lane = t & 31;
    const int half = lane >> 4;   // K pair select within fragment
    const int mn   = lane & 15;   // A: row M; D: col N

    const int tile = blockIdx.x * 8 + wave;   // one 16-atom tile per wave
    const int row0 = tile * 16;
    if (row0 >= n_atoms) return;              // wave-uniform

    const float* __restrict__ xrow = x + (long)(row0 + mn) * DTOT;
    const float* __restrict__ b0p  = lds + B0_OFF + lane * 2;
    const float* __restrict__ b1p  = lds + B1_OFF + lane * 2;
    const float* __restrict__ b2p  = lds + B2_OFF + lane * 2;

    // =================== y0: x[:, 0:128] @ w0 (K=128, N=128) ===================
    {
        const float s0 = 0.08838834764831845f;  // 1/sqrt(128)
        v8f acc[8] = {};
        #pragma unroll 4
        for (int kc = 0; kc < 32; ++kc) {
            const v2f a = *(const v2f*)(xrow + kc * 4 + 2 * half);
            #pragma unroll
            for (int nt = 0; nt < 8; ++nt) {
                const v2f b = *(const v2f*)(b0p + (kc * 8 + nt) * 64);
                acc[nt] = __builtin_amdgcn_wmma_f32_16x16x4_f32(
                    false, a, false, b, (short)0, acc[nt], false, false);
            }
        }
        #pragma unroll
        for (int nt = 0; nt < 8; ++nt)
            #pragma unroll
            for (int r = 0; r < 8; ++r)
                out[(long)(row0 + r + 8 * half) * DTOT + nt * 16 + mn] = acc[nt][r] * s0;
    }

    // ========== y1: 3 channels of (n x 64) @ w1, cols interleaved by 3 ==========
    {
        const float s1 = 0.125f;                // 1/sqrt(64)
        for (int i = 0; i < 3; ++i) {
            v8f acc[4] = {};
            #pragma unroll 4
            for (int kc = 0; kc < 16; ++kc) {
                const int k = kc * 4 + 2 * half;
                v2f a;
                a.x = xrow[D0 + 3 * k + i];
                a.y = xrow[D0 + 3 * (k + 1) + i];
                #pragma unroll
                for (int nt = 0; nt < 4; ++nt) {
                    const v2f b = *(const v2f*)(b1p + (kc * 4 + nt) * 64);
                    acc[nt] = __builtin_amdgcn_wmma_f32_16x16x4_f32(
                        false, a, false, b, (short)0, acc[nt], false, false);
                }
            }
            #pragma unroll
            for (int nt = 0; nt < 4; ++nt)
                #pragma unroll
                for (int r = 0; r < 8; ++r)
                    out[(long)(row0 + r + 8 * half) * DTOT + D0 + 3 * (nt * 16 + mn) + i] =
                        acc[nt][r] * s1;
        }
    }

    // ========== y2: 5 channels of (n x 32) @ w2, cols interleaved by 5 ==========
    {
        const float s2 = 0.17677669529663687f;  // 1/sqrt(32)
        for (int i = 0; i < 5; ++i) {
            v8f acc[2] = {};
            #pragma unroll 4
            for (int kc = 0; kc < 8; ++kc) {
                const int k = kc * 4 + 2 * half;
                v2f a;
                a.x = xrow[D0 + D1 + 5 * k + i];
                a.y = xrow[D0 + D1 + 5 * (k + 1) + i];
                #pragma unroll
                for (int nt = 0; nt < 2; ++nt) {
                    const v2f b = *(const v2f*)(b2p + (kc * 2 + nt) * 64);
                    acc[nt] = __builtin_amdgcn_wmma_f32_16x16x4_f32(
                        false, a, false, b, (short)0, acc[nt], false, false);
                }
            }
            #pragma unroll
            for (int nt = 0; nt < 2; ++nt)
                #pragma unroll
                for (int r = 0; r < 8; ++r)
                    out[(long)(row0 + r + 8 * half) * DTOT + D0 + D1 + 5 * (nt * 16 + mn) + i] =
                        acc[nt][r] * s2;
        }
    }
}

extern "C" void kernel_launch(void* const* d_in, const int* in_sizes, int n_in,
                              void* d_out, int out_size, void* d_ws, size_t ws_size,
                              hipStream_t stream) {
    const float* x  = (const float*)d_in[0];
    const float* w0 = (const float*)d_in[1];
    const float* w1 = (const float*)d_in[2];
    const float* w2 = (const float*)d_in[3];
    float* out = (float*)d_out;

    const int n_atoms = in_sizes[0] / DTOT;       // 262144
    const int tiles   = (n_atoms + 15) / 16;      // one 16-row tile per wave
    const int blocks  = (tiles + 7) / 8;          // 8 waves (256 thr) per block
    const size_t ldsBytes = (size_t)LDS_FLOATS * sizeof(float);

    sc_linear_wmma_f32<<<blocks, 256, ldsBytes, stream>>>(x, w0, w1, w2, out, n_atoms);
}